// CGCalculatorSingle_56624848830594
// MI455X (gfx1250) — compile-verified
//
#include <hip/hip_runtime.h>

// CG tensor contraction as batched 16x16 GEMM with K=128 against the
// (CG coefficient x mu-scatter) matrix, using V_WMMA_F32_16X16X4_F32.
// Bandwidth-bound (~68 MB -> ~3us at 23.3 TB/s). All data-dependent control
// flow (mu scatter, gather indices) is baked into small L2/WGP$-resident
// tables by a prep kernel so the hot loop is branch-free straight-line code.

typedef __attribute__((ext_vector_type(2))) float v2f;
typedef __attribute__((ext_vector_type(8))) float v8f;

#define NNZ 128
#define MVEC 11            // M1 == M2 == MU == 11
#define ROWS_PER_TILE 16
#define WAVES_PER_BLOCK 8
#define NCHUNK (NNZ / 4)   // 32 WMMA K-chunks

// Workspace layout:
//   Btab  : 1024 x float2  (8192 B)  -- B-matrix in WMMA lane order
//   offtab:   64 x uint    ( 256 B)  -- packed gather byte-offsets
#define OFFTAB_BYTEOFF 8192

// Build the WMMA-ready tables.
//   Btab[(kk*2+hi)*16 + n] = { (mu[ka]==n)?C[ka]:0, (mu[ka+1]==n)?C[ka+1]:0 }
//     with ka = 4*kk + 2*hi  (exactly what lane (n,hi) feeds the B operand).
//   offtab[kk*2+hi] = byte offsets {m1[ka],m2[ka],m1[ka+1],m2[ka+1]} * 4,
//     one per byte (offsets <= 40, fit in 8 bits).
__global__ void cg_prep_kernel(const int* __restrict__ m1,
                               const int* __restrict__ m2,
                               const int* __restrict__ mu,
                               const float* __restrict__ C,
                               float2* __restrict__ Btab,
                               unsigned* __restrict__ offtab) {
    const int t = threadIdx.x;                 // 256 threads
    for (int i = t; i < (NCHUNK * 2) * 16; i += 256) {
        const int n    = i & 15;
        const int pair = i >> 4;               // kk*2 + hi
        const int ka   = pair * 2;             // 4*kk + 2*hi
        float2 b;
        b.x = (mu[ka]     == n) ? C[ka]     : 0.0f;
        b.y = (mu[ka + 1] == n) ? C[ka + 1] : 0.0f;
        Btab[i] = b;
    }
    if (t < NCHUNK * 2) {
        const int ka = t * 2;
        const unsigned o = (unsigned)(m1[ka] * 4)
                         | ((unsigned)(m2[ka] * 4)     << 8)
                         | ((unsigned)(m1[ka + 1] * 4) << 16)
                         | ((unsigned)(m2[ka + 1] * 4) << 24);
        offtab[t] = o;
    }
}

__global__ __launch_bounds__(256) void cg_wmma_kernel(
    const float* __restrict__ X1,
    const float* __restrict__ X2,
    const float2*  __restrict__ Btab,
    const unsigned* __restrict__ offtab,
    float*       __restrict__ out,
    int nrows) {                       // nrows = B*F
    // Per-wave staging: [wave][X1|X2][row][m]
    __shared__ float lds[WAVES_PER_BLOCK][2][ROWS_PER_TILE][MVEC];

    const int tid  = threadIdx.x;
    const int lane = tid & 31;
    const int w    = tid >> 5;
    const int hi   = lane >> 4;        // 0: lanes 0-15, 1: lanes 16-31
    const int r    = lane & 15;        // A-matrix row within tile
    const int n    = lane & 15;        // B/D column for this lane

    int tile = blockIdx.x * WAVES_PER_BLOCK + w;
    const int ntiles = nrows / ROWS_PER_TILE;
    const bool active = (tile < ntiles);
    if (!active) tile = 0;             // clamp: keep wave alive through barrier/WMMA
    const int Rbase = tile * ROWS_PER_TILE;

    // Stage 16 rows of X1 (lanes 0-15) and X2 (lanes 16-31) into LDS.
    {
        const float* src  = hi ? X2 : X1;
        const float* rowp = src + (size_t)(Rbase + r) * MVEC;
        float* dst = &lds[w][hi][r][0];
        #pragma unroll
        for (int j = 0; j < MVEC; ++j) dst[j] = rowp[j];
    }
    __syncthreads();

    const char* x1b = (const char*)&lds[w][0][r][0];
    const char* x2b = (const char*)&lds[w][1][r][0];

    // D = A * B + D over 32 chunks of K=4; branch-free inner loop.
    v8f acc = {0.f, 0.f, 0.f, 0.f, 0.f, 0.f, 0.f, 0.f};
    #pragma unroll
    for (int kk = 0; kk < NCHUNK; ++kk) {
        const int pair   = kk * 2 + hi;
        const unsigned o = offtab[pair];              // broadcast per half-wave
        const float2 b   = Btab[pair * 16 + n];       // coalesced b64, WGP$-hot
        v2f A, Bv;
        A.x  = *(const float*)(x1b + (o & 255u))
             * *(const float*)(x2b + ((o >> 8) & 255u));
        A.y  = *(const float*)(x1b + ((o >> 16) & 255u))
             * *(const float*)(x2b + (o >> 24));
        Bv.x = b.x;
        Bv.y = b.y;
        // 8 args: (neg_a, A, neg_b, B, c_mod, C, reuse_a, reuse_b)
        acc = __builtin_amdgcn_wmma_f32_16x16x4_f32(
            false, A, false, Bv, (short)0, acc, false, false);
    }

    // C/D layout: lane l, VGPR v -> row (v + 8*hi), col n. Only n < 11 is real.
    if (active && n < MVEC) {
        #pragma unroll
        for (int v = 0; v < 8; ++v) {
            const int M = v + 8 * hi;
            out[(size_t)(Rbase + M) * MVEC + n] = acc[v];
        }
    }
}

extern "C" void kernel_launch(void* const* d_in, const int* in_sizes, int n_in,
                              void* d_out, int out_size, void* d_ws, size_t ws_size,
                              hipStream_t stream) {
    const float* X1 = (const float*)d_in[0];
    const float* X2 = (const float*)d_in[1];
    const int*   m1 = (const int*)d_in[2];
    const int*   m2 = (const int*)d_in[3];
    const int*   mu = (const int*)d_in[4];
    const float* C  = (const float*)d_in[5];
    float* out = (float*)d_out;

    float2*   Btab   = (float2*)d_ws;
    unsigned* offtab = (unsigned*)((char*)d_ws + OFFTAB_BYTEOFF);

    const int nrows  = in_sizes[0] / MVEC;            // B*F = 512000
    const int ntiles = nrows / ROWS_PER_TILE;         // 32000
    const int blocks = (ntiles + WAVES_PER_BLOCK - 1) / WAVES_PER_BLOCK; // 4000

    cg_prep_kernel<<<1, 256, 0, stream>>>(m1, m2, mu, C, Btab, offtab);
    cg_wmma_kernel<<<blocks, 32 * WAVES_PER_BLOCK, 0, stream>>>(
        X1, X2, Btab, offtab, out, nrows);
}